// GeometricProductDense_30442728194055
// MI455X (gfx1250) — compile-verified
//
#include <hip/hip_runtime.h>

typedef __attribute__((ext_vector_type(2))) float    v2f;
typedef __attribute__((ext_vector_type(8))) float    v8f;
typedef __attribute__((ext_vector_type(4))) unsigned v4u;
typedef __attribute__((ext_vector_type(8))) int      v8i;
typedef __attribute__((ext_vector_type(4))) int      v4i;

#define NB    8
#define NIN   32
#define KDIM  256           // reduction dim  (N_IN * 8 blades)
#define NDIM  256           // output dim     (UNITS * 8 blades)
#define KCHUNK 32
#define NCHUNKS (KDIM / KCHUNK)          // 8
#define ROWS_PER_BLK 128
// fragment-order W: [chunk][kstep(8)][tile(16)][lane(32)] -> float2
#define WFRAG_CHUNK_F2 (8 * 16 * 32)     // 4096 float2 = 32KB per chunk

// --------------------------------------------------------------------------
// Kernel 1: fold Cayley table into WMMA-fragment-ordered weights.
// Element e of float2 fid=[c][ks][t][lane]:
//   K   = c*32 + ks*4 + 2*(lane>>4) + e      (B-matrix row, = n*8+i)
//   col = t*16 + (lane&15)                   (B-matrix col, = u*8+k)
//   val = sign(i, i^k) * kernel[u][n][i^k]
// --------------------------------------------------------------------------
__global__ void ga_build_wfrag(const float* __restrict__ kernel,
                               float2* __restrict__ wfrag) {
    int fid  = blockIdx.x * blockDim.x + threadIdx.x;   // 0 .. 32767
    int lane = fid & 31;
    int t    = (fid >> 5) & 15;
    int ks   = (fid >> 9) & 7;
    int c    = fid >> 12;
    int col  = t * 16 + (lane & 15);
    int u = col >> 3, k = col & 7;
    float vals[2];
    #pragma unroll
    for (int e = 0; e < 2; ++e) {
        int K = c * KCHUNK + ks * 4 + 2 * (lane >> 4) + e;
        int n = K >> 3, i = K & 7;
        int j = i ^ k;
        int swaps = 0, b = i >> 1;
        while (b) { swaps += __popc(b & j); b >>= 1; }
        float s = (swaps & 1) ? -1.0f : 1.0f;
        vals[e] = s * kernel[(u * NIN + n) * NB + j];
    }
    wfrag[fid] = make_float2(vals[0], vals[1]);
}

// --------------------------------------------------------------------------
// Kernel 2: Out = relu(X[65536,256] @ W[256,256] + bias(k==0))
// 8 waves/block as 2x4; each wave owns a 64x64 tile (4x4 WMMA tiles).
// W chunk staged by TDM (tensor_load_to_lds), X chunk staged cooperatively
// into XOR-swizzled fragment order. All frag reads are single ds_load_b64.
// --------------------------------------------------------------------------
__global__ void __launch_bounds__(256, 1)
ga_gemm(const float* __restrict__ X,
        const float2* __restrict__ Wfrag,
        const float* __restrict__ bias,
        float* __restrict__ Out) {
    __shared__ float2 w_lds[WFRAG_CHUNK_F2];   // 32 KB
    __shared__ float2 x_lds[8 * 8 * 32];       // 16 KB: [strip][ks][lane^swz]

    const int tid    = threadIdx.x;
    const int wave   = tid >> 5;
    const int lane   = tid & 31;
    const int laneLo = lane & 15;
    const int laneHi = lane >> 4;
    const int mg     = wave >> 2;              // 0..1 : row group (64 rows)
    const int ng     = wave & 3;               // 0..3 : col group (64 cols)

    const long blockRow = (long)blockIdx.x * ROWS_PER_BLK;

    v8f acc[4][4] = {};

    for (int c = 0; c < NCHUNKS; ++c) {
        __syncthreads();

        // ---- W chunk: contiguous 32 KB -> LDS via Tensor Data Mover ----
#if __has_builtin(__builtin_amdgcn_tensor_load_to_lds)
        if (wave == 0) {
            unsigned long long ga =
                (unsigned long long)(const void*)(Wfrag + c * WFRAG_CHUNK_F2);
            unsigned ldsa = (unsigned)(unsigned long long)(const void*)&w_lds[0];
            // D# group 0: count=1 | lds_addr | global_addr[56:0] | type=2
            v4u g0 = { 1u,
                       ldsa,
                       (unsigned)(ga & 0xffffffffu),
                       (unsigned)((ga >> 32) & 0x01ffffffu) | (2u << 30) };
            // D# group 1: data_size=4B, tensor_dim0=65536, tensor_dim1=1,
            //             tile_dim0=8192, tile_dim1=1, dim0_stride=65536
            v8i g1 = { (int)(2u << 16),
                       0,
                       (int)((65536u >> 16) | (1u << 16)),
                       (int)(8192u << 16),
                       1,
                       65536,
                       0, 0 };
            v4i g2 = { 0, 0, 0, 0 };
            v4i g3 = { 0, 0, 0, 0 };
            v8i g4 = { 0, 0, 0, 0, 0, 0, 0, 0 };
            __builtin_amdgcn_tensor_load_to_lds(g0, g1, g2, g3, g4, 0);
        }
#else
        {
            const float4* src = (const float4*)(Wfrag + c * WFRAG_CHUNK_F2);
            float4* dst = (float4*)&w_lds[0];
            #pragma unroll
            for (int i = 0; i < 8; ++i) dst[tid + i * 256] = src[tid + i * 256];
        }
#endif

        // ---- X chunk: coalesced float4 reads, swizzled frag-order stores ----
        #pragma unroll
        for (int i = 0; i < 4; ++i) {
            int f   = tid + i * 256;           // 1024 float4: 128 rows x 8/row
            int row = f >> 3, kq = f & 7;
            float4 v = *(const float4*)(X + (blockRow + row) * KDIM
                                          + c * KCHUNK + kq * 4);
            int strip = row >> 4, r = row & 15;
            int base  = (strip * 8 + kq) * 32;
            int swz   = kq * 4;
            x_lds[base + ((r)      ^ swz)] = make_float2(v.x, v.y);  // K off 0,1
            x_lds[base + ((16 + r) ^ swz)] = make_float2(v.z, v.w);  // K off 2,3
        }

#if __has_builtin(__builtin_amdgcn_tensor_load_to_lds)
        if (wave == 0) __builtin_amdgcn_s_wait_tensorcnt(0);
#endif
        __syncthreads();

        if (c + 1 < NCHUNKS)   // nudge next X slice toward the caches
            __builtin_prefetch(X + (blockRow + (tid >> 3)) * KDIM
                                 + (c + 1) * KCHUNK, 0, 0);

        #pragma unroll
        for (int ks = 0; ks < KCHUNK / 4; ++ks) {
            v2f a[4], b[4];
            #pragma unroll
            for (int ss = 0; ss < 4; ++ss) {
                int strip = mg * 4 + ss;
                float2 av = x_lds[(strip * 8 + ks) * 32 + (lane ^ (ks * 4))];
                v2f t; t.x = av.x; t.y = av.y; a[ss] = t;
            }
            #pragma unroll
            for (int tt = 0; tt < 4; ++tt) {
                float2 bv = w_lds[(ks * 16 + ng * 4 + tt) * 32 + lane];
                v2f t; t.x = bv.x; t.y = bv.y; b[tt] = t;
            }
            #pragma unroll
            for (int ss = 0; ss < 4; ++ss)
                #pragma unroll
                for (int tt = 0; tt < 4; ++tt)
                    acc[ss][tt] = __builtin_amdgcn_wmma_f32_16x16x4_f32(
                        false, a[ss], false, b[tt], (short)0,
                        acc[ss][tt], false, false);
        }
    }

    // ---- epilogue: bias on scalar blade (col%8==0), ReLU, store ----
    #pragma unroll
    for (int ss = 0; ss < 4; ++ss) {
        const long rbase = blockRow + mg * 64 + ss * 16 + laneHi * 8;
        #pragma unroll
        for (int tt = 0; tt < 4; ++tt) {
            const int col = ng * 64 + tt * 16 + laneLo;
            const float bb = ((col & 7) == 0) ? bias[col >> 3] : 0.0f;
            float* outp = Out + rbase * NDIM + col;
            #pragma unroll
            for (int r = 0; r < 8; ++r) {      // M = r + 8*laneHi
                float v = acc[ss][tt][r] + bb;
                outp[(long)r * NDIM] = v > 0.0f ? v : 0.0f;
            }
        }
    }
}

// --------------------------------------------------------------------------
extern "C" void kernel_launch(void* const* d_in, const int* in_sizes, int n_in,
                              void* d_out, int out_size, void* d_ws, size_t ws_size,
                              hipStream_t stream) {
    (void)n_in; (void)out_size; (void)ws_size;
    const float* x      = (const float*)d_in[0];   // [B, 32, 8] == [B, 256]
    const float* kernel = (const float*)d_in[1];   // [32, 32, 8]
    const float* bias   = (const float*)d_in[2];   // [32, 1]
    float* out          = (float*)d_out;           // [B, 256]
    float2* wfrag       = (float2*)d_ws;           // 32768 float2 = 256 KB

    ga_build_wfrag<<<128, 256, 0, stream>>>(kernel, wfrag);

    const int Brows = in_sizes[0] / KDIM;          // 65536
    ga_gemm<<<Brows / ROWS_PER_BLK, 256, 0, stream>>>(x, wfrag, bias, out);
}